// RelationRNN_80762565034490
// MI455X (gfx1250) — compile-verified
//
#include <hip/hip_runtime.h>
#include <math.h>

#define E_EDGES  1000000
#define R_REL    1000
#define D_DIM    200
#define THREE_D  600
#define M_ROWS   2000      // 2*R rows: 0..999 = src, 1000..1999 = dst
#define LDSTRIDE 204       // padded LDS row stride (floats), conflict-free for 16 rows

typedef float v2f __attribute__((ext_vector_type(2)));
typedef float v8f __attribute__((ext_vector_type(8)));

// ---------------------------------------------------------------- zero scratch
__global__ void zero_ws(float* ws, int n) {
    int i = blockIdx.x * blockDim.x + threadIdx.x;
    if (i < n) ws[i] = 0.0f;
}

// ------------------------------------------------- scatter-sum over edges (L2-resident accumulator)
// One thread per (edge, 4-dim chunk): float4 gathers, f32 atomics into 1.6MB sum buffer.
__global__ void scatter_kernel(const int* __restrict__ ei,   // (2,E)
                               const int* __restrict__ et,   // (E)
                               const int* __restrict__ nid,  // (N_BATCH)
                               const float* __restrict__ emb,// (N_TOTAL, D)
                               float* __restrict__ sum,      // (2000, 200)
                               float* __restrict__ cnt) {    // (1000)
    long long idx   = (long long)blockIdx.x * blockDim.x + threadIdx.x;
    long long total = (long long)E_EDGES * (D_DIM / 4);
    if (idx >= total) return;
    int e  = (int)(idx / (D_DIM / 4));
    int d4 = (int)(idx - (long long)e * (D_DIM / 4)) * 4;
    int r = et[e];
    int s = nid[ei[e]];
    int t = nid[ei[E_EDGES + e]];
    const float4 vs = *(const float4*)(emb + (size_t)s * D_DIM + d4);
    const float4 vt = *(const float4*)(emb + (size_t)t * D_DIM + d4);
    float* ps = sum + (size_t)r * D_DIM + d4;
    float* pt = sum + (size_t)(R_REL + r) * D_DIM + d4;
    atomicAdd(ps + 0, vs.x); atomicAdd(ps + 1, vs.y);
    atomicAdd(ps + 2, vs.z); atomicAdd(ps + 3, vs.w);
    atomicAdd(pt + 0, vt.x); atomicAdd(pt + 1, vt.y);
    atomicAdd(pt + 2, vt.z); atomicAdd(pt + 3, vt.w);
    if (d4 == 0) atomicAdd(cnt + r, 1.0f);
}

// ---------------------------------------------------------------- mean (in place)
__global__ void mean_kernel(float* __restrict__ sum, const float* __restrict__ cnt) {
    int i = blockIdx.x * blockDim.x + threadIdx.x;
    if (i >= M_ROWS * D_DIM) return;
    int m = i / D_DIM;
    sum[i] = sum[i] / fmaxf(cnt[m % R_REL], 1.0f);
}

__device__ __forceinline__ float sigmoidf_(float x) { return 1.0f / (1.0f + expf(-x)); }

// ---------------------------------------------------------------- fused GRU via f32 WMMA
// One wave32 per 16 rows. A: 16x4 f32 (2 VGPR), B: 4x16 f32 (2 VGPR), C: 16x16 f32 (8 VGPR).
// A/B frag packing: VGPR0 -> K = 0 (lanes 0-15) / 2 (lanes 16-31); VGPR1 -> K = 1 / 3.
// C layout: VGPR i, lanes 0-15 -> M=i, lanes 16-31 -> M=8+i, N = lane%16.
__launch_bounds__(32)
__global__ void gru_wmma_kernel(const float* __restrict__ X,   // (2000, 200) means
                                const float* __restrict__ dyn, // (1000, 1, 200, 2) h-state
                                const float* __restrict__ Wih, // (600, 200)
                                const float* __restrict__ Whh, // (600, 200)
                                const float* __restrict__ bih, // (600)
                                const float* __restrict__ bhh, // (600)
                                float* __restrict__ out) {     // (1000, 1, 200, 2)
    __shared__ float Xs[16 * LDSTRIDE];
    __shared__ float Hs[16 * LDSTRIDE];
    const int lane = threadIdx.x;        // 0..31
    const int m0   = blockIdx.x * 16;

    // Stage X rows and the strided H rows (dyn[..., flag]) into LDS.
    for (int idx = lane; idx < 16 * D_DIM; idx += 32) {
        int row = idx / D_DIM, col = idx - row * D_DIM;
        int m = m0 + row;
        int r = m % R_REL, f = m / R_REL;
        Xs[row * LDSTRIDE + col] = X[(size_t)m * D_DIM + col];
        Hs[row * LDSTRIDE + col] = dyn[(size_t)r * (2 * D_DIM) + col * 2 + f];
    }
    __syncthreads();

    const int aRow = lane & 15;
    const int nRow = lane & 15;
    const int kOff = (lane < 16) ? 0 : 2;
    const int hi   = (lane < 16) ? 0 : 8;

    // -------- 12 full column tiles: cols 0..191, all W rows in-bounds (max 591) -> no guards.
    for (int c0 = 0; c0 < 192; c0 += 16) {
        const float* pI0 = Wih + (size_t)(c0 + nRow) * D_DIM + kOff;
        const float* pI1 = pI0 + (size_t)D_DIM * D_DIM;          // +200 W rows
        const float* pI2 = pI1 + (size_t)D_DIM * D_DIM;
        const float* pH0 = Whh + (size_t)(c0 + nRow) * D_DIM + kOff;
        const float* pH1 = pH0 + (size_t)D_DIM * D_DIM;
        const float* pH2 = pH1 + (size_t)D_DIM * D_DIM;

        v8f c_ir = {}; v8f c_iz = {}; v8f c_in = {};
        v8f c_hr = {}; v8f c_hz = {}; v8f c_hn = {};

        for (int k0 = 0; k0 < D_DIM; k0 += 4) {
            v2f ax, ah, b0, b1, b2, b3, b4, b5;
            ax.x = Xs[aRow * LDSTRIDE + k0 + kOff];
            ax.y = Xs[aRow * LDSTRIDE + k0 + kOff + 1];
            ah.x = Hs[aRow * LDSTRIDE + k0 + kOff];
            ah.y = Hs[aRow * LDSTRIDE + k0 + kOff + 1];
            b0.x = pI0[k0]; b0.y = pI0[k0 + 1];
            b1.x = pI1[k0]; b1.y = pI1[k0 + 1];
            b2.x = pI2[k0]; b2.y = pI2[k0 + 1];
            b3.x = pH0[k0]; b3.y = pH0[k0 + 1];
            b4.x = pH1[k0]; b4.y = pH1[k0 + 1];
            b5.x = pH2[k0]; b5.y = pH2[k0 + 1];

            c_ir = __builtin_amdgcn_wmma_f32_16x16x4_f32(false, ax, false, b0, (short)0, c_ir, false, false);
            c_iz = __builtin_amdgcn_wmma_f32_16x16x4_f32(false, ax, false, b1, (short)0, c_iz, false, false);
            c_in = __builtin_amdgcn_wmma_f32_16x16x4_f32(false, ax, false, b2, (short)0, c_in, false, false);
            c_hr = __builtin_amdgcn_wmma_f32_16x16x4_f32(false, ah, false, b3, (short)0, c_hr, false, false);
            c_hz = __builtin_amdgcn_wmma_f32_16x16x4_f32(false, ah, false, b4, (short)0, c_hz, false, false);
            c_hn = __builtin_amdgcn_wmma_f32_16x16x4_f32(false, ah, false, b5, (short)0, c_hn, false, false);
        }

        // Gate math on C-register layout, scatter to (R,1,D,2). All columns valid here.
        int dCol = c0 + nRow;
        float b_ir = bih[dCol], b_iz = bih[D_DIM + dCol], b_in = bih[2 * D_DIM + dCol];
        float b_hr = bhh[dCol], b_hz = bhh[D_DIM + dCol], b_hn = bhh[2 * D_DIM + dCol];
        #pragma unroll
        for (int i = 0; i < 8; ++i) {
            int   m  = m0 + hi + i;
            float h  = Hs[(hi + i) * LDSTRIDE + dCol];
            float rg = sigmoidf_(c_ir[i] + b_ir + c_hr[i] + b_hr);
            float zg = sigmoidf_(c_iz[i] + b_iz + c_hz[i] + b_hz);
            float ng = tanhf(c_in[i] + b_in + rg * (c_hn[i] + b_hn));
            float o  = (1.0f - zg) * ng + zg * h;
            int r = m % R_REL, f = m / R_REL;
            out[(size_t)r * (2 * D_DIM) + dCol * 2 + f] = o;
        }
    }

    // -------- Tail tile: c0 = 192, cols 192..207 (valid 192..199). Only the i_n/h_n
    // column group (W rows 592..607) can run out of bounds -> guard just those loads.
    {
        const int c0 = 192;
        const bool okN = (nRow < 8);                      // 592 + nRow < 600
        const float* pI0 = Wih + (size_t)(c0 + nRow) * D_DIM + kOff;
        const float* pI1 = pI0 + (size_t)D_DIM * D_DIM;
        const float* pI2 = pI1 + (size_t)D_DIM * D_DIM;
        const float* pH0 = Whh + (size_t)(c0 + nRow) * D_DIM + kOff;
        const float* pH1 = pH0 + (size_t)D_DIM * D_DIM;
        const float* pH2 = pH1 + (size_t)D_DIM * D_DIM;

        v8f c_ir = {}; v8f c_iz = {}; v8f c_in = {};
        v8f c_hr = {}; v8f c_hz = {}; v8f c_hn = {};

        for (int k0 = 0; k0 < D_DIM; k0 += 4) {
            v2f ax, ah, b0, b1, b2, b3, b4, b5;
            ax.x = Xs[aRow * LDSTRIDE + k0 + kOff];
            ax.y = Xs[aRow * LDSTRIDE + k0 + kOff + 1];
            ah.x = Hs[aRow * LDSTRIDE + k0 + kOff];
            ah.y = Hs[aRow * LDSTRIDE + k0 + kOff + 1];
            b0.x = pI0[k0]; b0.y = pI0[k0 + 1];
            b1.x = pI1[k0]; b1.y = pI1[k0 + 1];
            b2.x = okN ? pI2[k0] : 0.0f; b2.y = okN ? pI2[k0 + 1] : 0.0f;
            b3.x = pH0[k0]; b3.y = pH0[k0 + 1];
            b4.x = pH1[k0]; b4.y = pH1[k0 + 1];
            b5.x = okN ? pH2[k0] : 0.0f; b5.y = okN ? pH2[k0 + 1] : 0.0f;

            c_ir = __builtin_amdgcn_wmma_f32_16x16x4_f32(false, ax, false, b0, (short)0, c_ir, false, false);
            c_iz = __builtin_amdgcn_wmma_f32_16x16x4_f32(false, ax, false, b1, (short)0, c_iz, false, false);
            c_in = __builtin_amdgcn_wmma_f32_16x16x4_f32(false, ax, false, b2, (short)0, c_in, false, false);
            c_hr = __builtin_amdgcn_wmma_f32_16x16x4_f32(false, ah, false, b3, (short)0, c_hr, false, false);
            c_hz = __builtin_amdgcn_wmma_f32_16x16x4_f32(false, ah, false, b4, (short)0, c_hz, false, false);
            c_hn = __builtin_amdgcn_wmma_f32_16x16x4_f32(false, ah, false, b5, (short)0, c_hn, false, false);
        }

        int  dCol  = c0 + nRow;
        bool valid = dCol < D_DIM;                        // cols 192..199 only
        float b_ir = 0, b_iz = 0, b_in = 0, b_hr = 0, b_hz = 0, b_hn = 0;
        if (valid) {
            b_ir = bih[dCol]; b_iz = bih[D_DIM + dCol]; b_in = bih[2 * D_DIM + dCol];
            b_hr = bhh[dCol]; b_hz = bhh[D_DIM + dCol]; b_hn = bhh[2 * D_DIM + dCol];
        }
        #pragma unroll
        for (int i = 0; i < 8; ++i) {
            if (valid) {
                int   m  = m0 + hi + i;
                float h  = Hs[(hi + i) * LDSTRIDE + dCol];
                float rg = sigmoidf_(c_ir[i] + b_ir + c_hr[i] + b_hr);
                float zg = sigmoidf_(c_iz[i] + b_iz + c_hz[i] + b_hz);
                float ng = tanhf(c_in[i] + b_in + rg * (c_hn[i] + b_hn));
                float o  = (1.0f - zg) * ng + zg * h;
                int r = m % R_REL, f = m / R_REL;
                out[(size_t)r * (2 * D_DIM) + dCol * 2 + f] = o;
            }
        }
    }
}

extern "C" void kernel_launch(void* const* d_in, const int* in_sizes, int n_in,
                              void* d_out, int out_size, void* d_ws, size_t ws_size,
                              hipStream_t stream) {
    const int*   ei  = (const int*)d_in[0];   // edge_index (2,E)
    const int*   et  = (const int*)d_in[1];   // edge_type (E)
    const int*   nid = (const int*)d_in[2];   // node_id (N_BATCH)
    const float* emb = (const float*)d_in[3]; // static_entity_emb (N_TOTAL,D)
    const float* dyn = (const float*)d_in[4]; // dynamic_relation_emb (R,1,D,2)
    const float* Wih = (const float*)d_in[5];
    const float* Whh = (const float*)d_in[6];
    const float* bih = (const float*)d_in[7];
    const float* bhh = (const float*)d_in[8];
    float* out = (float*)d_out;

    float* sum = (float*)d_ws;                       // (2000, 200) f32
    float* cnt = sum + (size_t)M_ROWS * D_DIM;       // (1000) f32

    int nZero = M_ROWS * D_DIM + R_REL;
    zero_ws<<<(nZero + 255) / 256, 256, 0, stream>>>(sum, nZero);

    long long total = (long long)E_EDGES * (D_DIM / 4);
    int blocks = (int)((total + 255) / 256);
    scatter_kernel<<<blocks, 256, 0, stream>>>(ei, et, nid, emb, sum, cnt);

    mean_kernel<<<(M_ROWS * D_DIM + 255) / 256, 256, 0, stream>>>(sum, cnt);

    gru_wmma_kernel<<<M_ROWS / 16, 32, 0, stream>>>(sum, dyn, Wih, Whh, bih, bhh, out);
}